// LightGCN_68504728371439
// MI455X (gfx1250) — compile-verified
//
#include <hip/hip_runtime.h>

#define N_USERS 100000
#define N_ITEMS 50000
#define N_NODES (N_USERS + N_ITEMS)
#define N_EDGES 3000000
#define DIM 64
#define EDGES_PER_BLOCK 256

#define NELEM4 ((N_NODES * DIM) / 4)  // 2,400,000 float4 elements
#define USER4 ((N_USERS * DIM) / 4)   // 1,600,000 float4 elements

typedef __attribute__((address_space(1))) int g_int;    // global (AS1) int
typedef __attribute__((address_space(3))) int lds_int;  // LDS (AS3) int

__device__ __forceinline__ void atomic_fadd(float* p, float v) {
  // Relaxed, device-scope f32 add, result unused -> no-return
  // global_atomic_add_f32 (L2-side RMW).
  (void)__hip_atomic_fetch_add(p, v, __ATOMIC_RELAXED, __HIP_MEMORY_SCOPE_AGENT);
}

// Edge-parallel COO scatter: y[rows[e]] += vals[e] * x[cols[e]]
// 16 lanes per edge, float4 per lane (64 dims). Edge triples are staged to LDS
// via gfx1250 async global->LDS copies (ASYNCcnt path).
// NOTE: kept FIRST in this file so the disasm snippet shows this hot kernel.
__global__ __launch_bounds__(256) void scatter_kernel(
    const int* __restrict__ rows, const int* __restrict__ cols,
    const float* __restrict__ vals, const float* __restrict__ x,
    float* __restrict__ y) {
  __shared__ int   s_row[EDGES_PER_BLOCK];
  __shared__ int   s_col[EDGES_PER_BLOCK];
  __shared__ float s_val[EDGES_PER_BLOCK];

  const int tid  = threadIdx.x;
  const int base = blockIdx.x * EDGES_PER_BLOCK;
  const int ge   = base + tid;

#if defined(__gfx1250__)
  // Near-scope speculative prefetch of the edge stream ahead of this block
  // (OOB is harmless: speculative prefetches with failed translation drop).
  __builtin_prefetch(rows + base + 4 * EDGES_PER_BLOCK, 0, 3);
  __builtin_prefetch(cols + base + 4 * EDGES_PER_BLOCK, 0, 3);
  __builtin_prefetch(vals + base + 4 * EDGES_PER_BLOCK, 0, 3);
#endif

  if (ge < N_EDGES) {
#if defined(__gfx1250__) && __has_builtin(__builtin_amdgcn_global_load_async_to_lds_b32)
    __builtin_amdgcn_global_load_async_to_lds_b32(
        (g_int*)(rows + ge), (lds_int*)&s_row[tid], 0, 0);
    __builtin_amdgcn_global_load_async_to_lds_b32(
        (g_int*)(cols + ge), (lds_int*)&s_col[tid], 0, 0);
    __builtin_amdgcn_global_load_async_to_lds_b32(
        (g_int*)(vals + ge), (lds_int*)&s_val[tid], 0, 0);
#else
    s_row[tid] = rows[ge];
    s_col[tid] = cols[ge];
    s_val[tid] = vals[ge];
#endif
  }
#if defined(__gfx1250__)
#if __has_builtin(__builtin_amdgcn_s_wait_asynccnt)
  __builtin_amdgcn_s_wait_asynccnt(0);
#else
  asm volatile("s_wait_asynccnt 0" ::: "memory");
#endif
#endif
  __syncthreads();

  const int sub  = tid & 15;  // 4-dim chunk within the 64-dim embedding
  const int eoff = tid >> 4;  // edge slot within a 16-edge pass

  if (base + EDGES_PER_BLOCK <= N_EDGES) {
    // Full block (all but the last): no bounds checks, deep unroll for MLP.
#pragma unroll 8
    for (int p = 0; p < EDGES_PER_BLOCK / 16; ++p) {
      const int le = p * 16 + eoff;
      const int   row = s_row[le];
      const int   col = s_col[le];
      const float v   = s_val[le];
      const float4 g = *(const float4*)(x + (long long)col * DIM + sub * 4);
      float* dst = y + (long long)row * DIM + sub * 4;
      atomic_fadd(dst + 0, v * g.x);
      atomic_fadd(dst + 1, v * g.y);
      atomic_fadd(dst + 2, v * g.z);
      atomic_fadd(dst + 3, v * g.w);
    }
  } else {
    for (int p = 0; p < EDGES_PER_BLOCK / 16; ++p) {
      const int le = p * 16 + eoff;
      if (base + le >= N_EDGES) break;
      const int   row = s_row[le];
      const int   col = s_col[le];
      const float v   = s_val[le];
      const float4 g = *(const float4*)(x + (long long)col * DIM + sub * 4);
      float* dst = y + (long long)row * DIM + sub * 4;
      atomic_fadd(dst + 0, v * g.x);
      atomic_fadd(dst + 1, v * g.y);
      atomic_fadd(dst + 2, v * g.z);
      atomic_fadd(dst + 3, v * g.w);
    }
  }
}

// out4[i] = cur4[i] = concat(user,item)4[i]; nxt4[i] = 0
__global__ __launch_bounds__(256) void init_kernel(
    const float4* __restrict__ user_emb, const float4* __restrict__ item_emb,
    float4* __restrict__ out, float4* __restrict__ cur,
    float4* __restrict__ nxt) {
  int i = blockIdx.x * 256 + threadIdx.x;
  if (i >= NELEM4) return;
  float4 v = (i < USER4) ? user_emb[i] : item_emb[i - USER4];
  out[i] = v;
  cur[i] = v;
  nxt[i] = make_float4(0.f, 0.f, 0.f, 0.f);
}

// out4[i] += src4[i]; other4[i] = 0  (prepares next scatter target)
__global__ __launch_bounds__(256) void add_zero_kernel(
    float4* __restrict__ out, const float4* __restrict__ src,
    float4* __restrict__ other) {
  int i = blockIdx.x * 256 + threadIdx.x;
  if (i >= NELEM4) return;
  float4 o = out[i], s = src[i];
  o.x += s.x; o.y += s.y; o.z += s.z; o.w += s.w;
  out[i] = o;
  other[i] = make_float4(0.f, 0.f, 0.f, 0.f);
}

// out4[i] = (out4[i] + src4[i]) * 0.25
__global__ __launch_bounds__(256) void final_kernel(
    float4* __restrict__ out, const float4* __restrict__ src) {
  int i = blockIdx.x * 256 + threadIdx.x;
  if (i >= NELEM4) return;
  float4 o = out[i], s = src[i];
  o.x = (o.x + s.x) * 0.25f;
  o.y = (o.y + s.y) * 0.25f;
  o.z = (o.z + s.z) * 0.25f;
  o.w = (o.w + s.w) * 0.25f;
  out[i] = o;
}

extern "C" void kernel_launch(void* const* d_in, const int* in_sizes, int n_in,
                              void* d_out, int out_size, void* d_ws, size_t ws_size,
                              hipStream_t stream) {
  const float* user_emb = (const float*)d_in[0];
  const float* item_emb = (const float*)d_in[1];
  const int*   rows     = (const int*)d_in[2];
  const int*   cols     = (const int*)d_in[3];
  const float* vals     = (const float*)d_in[4];

  float* out  = (float*)d_out;
  float* buf0 = (float*)d_ws;                         // 38.4 MB
  float* buf1 = buf0 + (size_t)N_NODES * DIM;         // 38.4 MB

  dim3 eb(256), eg((NELEM4 + 255) / 256);             // 9375 blocks
  dim3 sb(256), sg((N_EDGES + EDGES_PER_BLOCK - 1) / EDGES_PER_BLOCK);

  // out = cur0 = emb; buf1 = 0
  init_kernel<<<eg, eb, 0, stream>>>((const float4*)user_emb,
                                     (const float4*)item_emb, (float4*)out,
                                     (float4*)buf0, (float4*)buf1);
  // layer 1: buf1 = A*buf0 ; out += buf1 ; buf0 = 0
  scatter_kernel<<<sg, sb, 0, stream>>>(rows, cols, vals, buf0, buf1);
  add_zero_kernel<<<eg, eb, 0, stream>>>((float4*)out, (const float4*)buf1,
                                         (float4*)buf0);
  // layer 2: buf0 = A*buf1 ; out += buf0 ; buf1 = 0
  scatter_kernel<<<sg, sb, 0, stream>>>(rows, cols, vals, buf1, buf0);
  add_zero_kernel<<<eg, eb, 0, stream>>>((float4*)out, (const float4*)buf0,
                                         (float4*)buf1);
  // layer 3: buf1 = A*buf0 ; out = (out + buf1) / 4
  scatter_kernel<<<sg, sb, 0, stream>>>(rows, cols, vals, buf0, buf1);
  final_kernel<<<eg, eb, 0, stream>>>((float4*)out, (const float4*)buf1);
}